// Yolov5_70660801954483
// MI455X (gfx1250) — compile-verified
//
#include <hip/hip_runtime.h>
#include <hip/hip_bf16.h>
#include <stdint.h>

// ---- static pipeline constants (match reference) ----
#define NANCH   15120
#define NCLS    3
#define PRED_C  (5 + NCLS)          // 8
#define KTOP    1000
#define KPAD    1024
#define SORTN   16384
#define CONF_T  0.25f
#define IOU_T   0.45f
#define MAXWH   4096.0f
#define PADW    0.0f
#define PADH    12.0f
#define INV_GAIN 2.0f               // 1/0.5
#define LIMW    1280.0f
#define LIMH    720.0f

typedef __attribute__((ext_vector_type(2))) float v2f;
typedef __attribute__((ext_vector_type(8))) float v8f;

#if __has_builtin(__builtin_amdgcn_global_load_async_to_lds_b32) && \
    __has_builtin(__builtin_amdgcn_s_wait_asynccnt)
#define USE_ASYNC_LDS 1
typedef __attribute__((address_space(1))) int gas_int;   // global (AS1)
typedef __attribute__((address_space(3))) int las_int;   // LDS (AS3)
#else
#define USE_ASYNC_LDS 0
#endif

#if __has_builtin(__builtin_amdgcn_wmma_f32_16x16x4_f32)
#define USE_WMMA 1
#else
#define USE_WMMA 0
#endif

__device__ __forceinline__ unsigned ballot32(bool p) {
#if __has_builtin(__builtin_amdgcn_ballot_w32)
  return __builtin_amdgcn_ballot_w32(p);
#else
  return (unsigned)__ballot((int)p);
#endif
}

__device__ __forceinline__ int uniform_i(int x) {
#if __has_builtin(__builtin_amdgcn_readfirstlane)
  return __builtin_amdgcn_readfirstlane(x);
#else
  return x;
#endif
}

// ---------------------------------------------------------------------------
// Kernel 1: decode. Stage 256 anchors (256*8 dwords) into LDS with async
// global->LDS copies (ASYNCcnt path), then decode xywh->xyxy + best class.
// ---------------------------------------------------------------------------
__global__ __launch_bounds__(256) void decode_kernel(
    const float* __restrict__ pred, float* __restrict__ boxesAll,
    float* __restrict__ confAll, int* __restrict__ clsAll) {
  const int bpi = (NANCH + 255) / 256;
  const int img = blockIdx.x / bpi;
  const int blk = blockIdx.x % bpi;
  const int a0  = blk * 256;
  const int cnt = min(256, NANCH - a0);
  const int t   = threadIdx.x;

  __shared__ float tile[256 * PRED_C];
  const float* gbase = pred + ((size_t)img * NANCH + a0) * PRED_C;
  const int ndw = cnt * PRED_C;

#if USE_ASYNC_LDS
  #pragma unroll
  for (int c = 0; c < PRED_C; ++c) {
    int d = c * 256 + t;
    if (d < ndw) {
      __builtin_amdgcn_global_load_async_to_lds_b32(
          (gas_int*)(gbase + d), (las_int*)(&tile[d]), 0, 0);
    }
  }
  __builtin_amdgcn_s_wait_asynccnt(0);
#else
  #pragma unroll
  for (int c = 0; c < PRED_C; ++c) {
    int d = c * 256 + t;
    if (d < ndw) tile[d] = gbase[d];
  }
#endif
  __syncthreads();

  if (t < cnt) {
    const float* p = &tile[t * PRED_C];
    float cx = p[0], cy = p[1], w2 = p[2] * 0.5f, h2 = p[3] * 0.5f;
    float obj = p[4];
    float best = -1.0f; int bc = 0;
    #pragma unroll
    for (int c = 0; c < NCLS; ++c) {
      float s = obj * p[5 + c];
      if (s > best) { best = s; bc = c; }
    }
    float cf = (best > CONF_T) ? best : 0.0f;
    size_t o = (size_t)img * NANCH + a0 + t;
    boxesAll[o * 4 + 0] = cx - w2;
    boxesAll[o * 4 + 1] = cy - h2;
    boxesAll[o * 4 + 2] = cx + w2;
    boxesAll[o * 4 + 3] = cy + h2;
    confAll[o] = cf;
    clsAll[o]  = bc;
  }
}

// ---------------------------------------------------------------------------
// Kernel 2: per-image descending bitonic sort of packed keys in 64KB LDS.
// key = (confbits & ~0x3FFF) | (0x3FFF - idx)  -> desc conf, asc index.
// ---------------------------------------------------------------------------
__global__ __launch_bounds__(1024) void topk_sort_kernel(
    const float* __restrict__ confAll, const float* __restrict__ boxesAll,
    const int* __restrict__ clsAll, float* __restrict__ topbox,
    float* __restrict__ topscore, int* __restrict__ topcls) {
  const int img = blockIdx.x;
  const int tid = threadIdx.x;
  __shared__ unsigned keys[SORTN];   // 64 KB

  for (int i = tid; i < SORTN; i += 1024) {
    unsigned key = 0u;
    if (i < NANCH) {
      unsigned cb = __float_as_uint(confAll[(size_t)img * NANCH + i]);
      key = (cb & ~0x3FFFu) | (0x3FFFu - (unsigned)i);
    }
    keys[i] = key;
  }
  __syncthreads();

  for (unsigned k = 2; k <= SORTN; k <<= 1) {
    for (unsigned j = k >> 1; j > 0; j >>= 1) {
      for (unsigned i = tid; i < SORTN; i += 1024) {
        unsigned ixj = i ^ j;
        if (ixj > i) {
          unsigned a = keys[i], b = keys[ixj];
          bool desc = ((i & k) == 0);
          if (desc ? (a < b) : (a > b)) { keys[i] = b; keys[ixj] = a; }
        }
      }
      __syncthreads();
    }
  }

  // write top KPAD entries (zeros past KTOP / below threshold)
  unsigned key = keys[tid];
  unsigned idx = 0x3FFFu - (key & 0x3FFFu);
  float sc = 0.0f, b0 = 0.0f, b1 = 0.0f, b2 = 0.0f, b3 = 0.0f;
  int c = 0;
  if (tid < KTOP && (key >> 14) != 0u && idx < NANCH) {
    size_t o = (size_t)img * NANCH + idx;
    sc = confAll[o];
    b0 = boxesAll[o * 4 + 0]; b1 = boxesAll[o * 4 + 1];
    b2 = boxesAll[o * 4 + 2]; b3 = boxesAll[o * 4 + 3];
    c  = clsAll[o];
  }
  size_t q = (size_t)img * KPAD + tid;
  topbox[q * 4 + 0] = b0; topbox[q * 4 + 1] = b1;
  topbox[q * 4 + 2] = b2; topbox[q * 4 + 3] = b3;
  topscore[q] = sc;
  topcls[q]   = c;
}

// ---------------------------------------------------------------------------
// Kernel 3: suppression bit matrix. One block per image, 32 waves.
// Wave `w` owns word-column w for ALL 64 row tiles, so the B fragment
// (ones / area_j) and the j-side box coords are loop-invariant. Per ti:
// 1 LDS area load + 2 WMMA f32 16x16x4 outer-sums (denominators
// area_i + area_j) + VALU intersections + division-free threshold test
// (inter > t*denom, denom > 0) + ballot bit-pack + unique-writer stores.
// ---------------------------------------------------------------------------
__global__ __launch_bounds__(1024) void iou_bits_kernel(
    const float* __restrict__ topbox, const float* __restrict__ topscore,
    const int* __restrict__ topcls, unsigned* __restrict__ suppG) {
  const int img  = blockIdx.x;
  const int tid  = threadIdx.x;
  const int wave = uniform_i(tid >> 5);   // wave-uniform word column
  const int lane = tid & 31;
  const int hs   = lane >> 4;      // half-select: 0 => rows v, 1 => rows v+8
  const int l16  = lane & 15;

  __shared__ float bx1[KPAD], by1[KPAD], bx2[KPAD], by2[KPAD], area[KPAD];

  {
    size_t q = (size_t)img * KPAD + tid;
    float x1 = topbox[q * 4 + 0], y1 = topbox[q * 4 + 1];
    float x2 = topbox[q * 4 + 2], y2 = topbox[q * 4 + 3];
    float off = (float)topcls[q] * MAXWH;       // class-offset trick
    bx1[tid] = x1 + off; by1[tid] = y1 + off;
    bx2[tid] = x2 + off; by2[tid] = y2 + off;
    area[tid] = (x2 - x1) * (y2 - y1);
    (void)topscore;
  }
  __syncthreads();

  unsigned* srow = suppG + (size_t)img * KPAD * 32;

  // loop-invariant j-side data for this wave's two j tiles (tj = 2w, 2w+1)
  int jj[2];
  float jx1[2], jy1[2], jx2[2], jy2[2];
  v2f bfrag[2];
  #pragma unroll
  for (int s = 0; s < 2; ++s) {
    int tj = 2 * wave + s;
    int j  = tj * 16 + l16;
    jj[s]  = j;
    jx1[s] = bx1[j]; jy1[s] = by1[j];
    jx2[s] = bx2[j]; jy2[s] = by2[j];
    float aj = area[j];                 // unconditional load, then select
    bfrag[s].x = hs ? 0.0f : 1.0f;      // row K0 = ones
    bfrag[s].y = hs ? 0.0f : aj;        // row K1 = area_j
  }

  for (int ti = 0; ti < KPAD / 16; ++ti) {
    // A fragment (16x4, f32): col K0 = area_i, col K1 = ones, K2/K3 = 0
    float ai = area[ti * 16 + l16];     // unconditional load, then select
    v2f afrag;
    afrag.x = hs ? 0.0f : ai;
    afrag.y = hs ? 0.0f : 1.0f;

    v8f dsum[2];
    #pragma unroll
    for (int s = 0; s < 2; ++s) {
#if USE_WMMA
      v8f cz = {};
      dsum[s] = __builtin_amdgcn_wmma_f32_16x16x4_f32(
          false, afrag, false, bfrag[s], (short)0, cz, false, false);
#else
      #pragma unroll
      for (int v = 0; v < 8; ++v)
        dsum[s][v] = area[ti * 16 + v + hs * 8] + area[jj[s]];
#endif
    }

    #pragma unroll
    for (int v = 0; v < 8; ++v) {
      int i = ti * 16 + v + hs * 8;
      float ax1 = bx1[i], ay1 = by1[i], ax2 = bx2[i], ay2 = by2[i];
      unsigned bm[2];
      #pragma unroll
      for (int s = 0; s < 2; ++s) {
        float ix1 = fmaxf(ax1, jx1[s]);
        float iy1 = fmaxf(ay1, jy1[s]);
        float ix2 = fminf(ax2, jx2[s]);
        float iy2 = fminf(ay2, jy2[s]);
        float iw = fmaxf(ix2 - ix1, 0.0f);
        float ih = fmaxf(iy2 - iy1, 0.0f);
        float inter = iw * ih;
        // iou > t  <=>  inter > t * (union + eps)   (union + eps > 0)
        float denom = dsum[s][v] - inter + 1e-7f;
        bm[s] = ballot32((inter > IOU_T * denom) && (jj[s] > i));
      }
      // lanes 0-15 carry row (ti*16+v); lanes 16-31 carry row (ti*16+v+8)
      unsigned word_lo = (bm[0] & 0xFFFFu) | (bm[1] << 16);
      unsigned word_hi = (bm[0] >> 16) | (bm[1] & 0xFFFF0000u);
      if (lane == 0)  srow[(size_t)(ti * 16 + v) * 32 + wave] = word_lo;
      if (lane == 16) srow[(size_t)(ti * 16 + v + 8) * 32 + wave] = word_hi;
    }
  }
}

// ---------------------------------------------------------------------------
// Kernel 4: greedy scan (single wave per image; keep mask lives in registers,
// one 32-bit word per lane; broadcast bit-test via ds_bpermute), then the
// scale_coords + output write.
// ---------------------------------------------------------------------------
__global__ __launch_bounds__(32) void nms_scan_kernel(
    const float* __restrict__ topbox, const float* __restrict__ topscore,
    const int* __restrict__ topcls, const unsigned* __restrict__ suppG,
    float* __restrict__ outBoxes, float* __restrict__ outScores,
    int* __restrict__ outLabels, int* __restrict__ outValid) {
  const int img  = blockIdx.x;
  const int lane = threadIdx.x;
  __shared__ unsigned kws[32];

  // keep0 = scores > 0; lane l owns bits [l*32, l*32+31]
  unsigned kw = 0u;
  for (int b = 0; b < 32; ++b) {
    float s = topscore[(size_t)img * KPAD + lane * 32 + b];
    kw |= (s > 0.0f ? 1u : 0u) << b;
  }

  const unsigned* srow = suppG + (size_t)img * KPAD * 32;
  for (int i = 0; i < KTOP; ++i) {
    unsigned wrd = (unsigned)__builtin_amdgcn_ds_bpermute((i >> 5) << 2, (int)kw);
    if ((wrd >> (i & 31)) & 1u) {          // alive -> suppress its victims
      kw &= ~srow[(size_t)i * 32 + lane];
    }
  }
  kws[lane] = kw;
  __syncthreads();

  for (int t = lane; t < KTOP; t += 32) {
    bool keep = (kws[t >> 5] >> (t & 31)) & 1u;
    size_t q = (size_t)img * KPAD + t;
    float x1 = topbox[q * 4 + 0], y1 = topbox[q * 4 + 1];
    float x2 = topbox[q * 4 + 2], y2 = topbox[q * 4 + 3];
    // scale_coords: (box - pad) / gain, clip, round (RTE matches jnp.round)
    x1 = rintf(fminf(fmaxf((x1 - PADW) * INV_GAIN, 0.0f), LIMW));
    y1 = rintf(fminf(fmaxf((y1 - PADH) * INV_GAIN, 0.0f), LIMH));
    x2 = rintf(fminf(fmaxf((x2 - PADW) * INV_GAIN, 0.0f), LIMW));
    y2 = rintf(fminf(fmaxf((y2 - PADH) * INV_GAIN, 0.0f), LIMH));
    float kf = keep ? 1.0f : 0.0f;
    size_t o = (size_t)img * KTOP + t;
    outBoxes[o * 4 + 0] = x1 * kf;
    outBoxes[o * 4 + 1] = y1 * kf;
    outBoxes[o * 4 + 2] = x2 * kf;
    outBoxes[o * 4 + 3] = y2 * kf;
    outScores[o] = topscore[q] * kf;
    outLabels[o] = keep ? (topcls[q] + 1) : 0;
    outValid[o]  = keep ? 1 : 0;
  }
}

// ---------------------------------------------------------------------------
extern "C" void kernel_launch(void* const* d_in, const int* in_sizes, int n_in,
                              void* d_out, int out_size, void* d_ws, size_t ws_size,
                              hipStream_t stream) {
  const float* pred = (const float*)d_in[0];
  int Btot = in_sizes[0] / (NANCH * PRED_C);
  if (Btot <= 0) Btot = 128;

  char* w = (char*)d_ws;
  auto take = [&](size_t bytes) -> char* {
    char* p = w;
    w += (bytes + 255) & ~(size_t)255;
    return p;
  };
  float*    boxesAll = (float*)   take((size_t)Btot * NANCH * 4 * sizeof(float));
  float*    confAll  = (float*)   take((size_t)Btot * NANCH * sizeof(float));
  int*      clsAll   = (int*)     take((size_t)Btot * NANCH * sizeof(int));
  float*    topbox   = (float*)   take((size_t)Btot * KPAD * 4 * sizeof(float));
  float*    topscore = (float*)   take((size_t)Btot * KPAD * sizeof(float));
  int*      topcls   = (int*)     take((size_t)Btot * KPAD * sizeof(int));
  unsigned* suppG    = (unsigned*)take((size_t)Btot * KPAD * 32 * sizeof(unsigned));
  (void)ws_size; (void)n_in; (void)out_size;

  const int bpi = (NANCH + 255) / 256;
  decode_kernel<<<dim3(Btot * bpi), dim3(256), 0, stream>>>(
      pred, boxesAll, confAll, clsAll);
  topk_sort_kernel<<<dim3(Btot), dim3(1024), 0, stream>>>(
      confAll, boxesAll, clsAll, topbox, topscore, topcls);
  iou_bits_kernel<<<dim3(Btot), dim3(1024), 0, stream>>>(
      topbox, topscore, topcls, suppG);

  float* fout = (float*)d_out;
  int*   iout = (int*)d_out;
  size_t OS = (size_t)Btot * KTOP * 4;          // scores offset (elements)
  size_t OL = OS + (size_t)Btot * KTOP;         // labels offset
  size_t OV = OL + (size_t)Btot * KTOP;         // valid offset
  nms_scan_kernel<<<dim3(Btot), dim3(32), 0, stream>>>(
      topbox, topscore, topcls, suppG,
      fout, fout + OS, iout + OL, iout + OV);
}